// GINEncoder_75514114998654
// MI455X (gfx1250) — compile-verified
//
#include <hip/hip_runtime.h>
#include <hip/hip_bf16.h>

#define D        128
#define LDSP     132          // padded LDS row stride: 132 mod 64 = 4 -> conflict-free
#define N_NODES  50000
#define N_EDGES  800000
#define BN_EPS   1e-5f

typedef __attribute__((ext_vector_type(2))) float v2f;
typedef __attribute__((ext_vector_type(8))) float v8f;

// Fused GEMM (M x 128 @ 128 x 128) + bias + optional eval-BN + ReLU [+ BN3 + ReLU].
// mode 0: out = relu(acc + bias)                       (input encoder)
// mode 1: out = relu(bn(acc + bias))                   (conv MLP first linear)
// mode 2: out = relu(bn3(relu(bn(acc + bias))))        (conv MLP second linear + outer BN)
// Block = 256 threads = 8 waves; block owns a 16-row M tile, wave w owns N tile [16w,16w+16).
__global__ __launch_bounds__(256) void gemm_bn_relu(
    const float* __restrict__ A, const float* __restrict__ W,
    const float* __restrict__ bias, const float* __restrict__ bn,
    const float* __restrict__ bn3, float* __restrict__ Out, int mode)
{
    __shared__ float As[16 * LDSP];   // 16x128 A tile, padded
    const int tid  = threadIdx.x;
    const int wave = tid >> 5;
    const int lane = tid & 31;
    const int m0   = blockIdx.x << 4;

    // Cooperative stage of the A tile into LDS (512 float4s, 2 per thread).
    {
        const float4* Ag = (const float4*)(A + (size_t)m0 * D);
        #pragma unroll
        for (int it = 0; it < 2; ++it) {
            int idx = tid + it * 256;              // 0..511
            int r = idx >> 5, q = idx & 31;        // row, float4-column
            *(float4*)(&As[r * LDSP + q * 4]) = Ag[idx];
        }
    }
    __syncthreads();

    const int n0   = wave << 4;
    const int half = lane >> 4;       // lane group selects K pair (A/B) / row+8 (C)
    const int l15  = lane & 15;

    // A frag per ISA layout: lane l15 -> row l15; vgpr0=K(k0+2*half), vgpr1=K(k0+2*half+1)
    const float* Arow = &As[l15 * LDSP + 2 * half];
    // B frag: lane l15 -> col n0+l15; rows k0+2*half and k0+2*half+1
    const float* Bcol = W + (size_t)(2 * half) * D + n0 + l15;

    v8f acc = {0.f, 0.f, 0.f, 0.f, 0.f, 0.f, 0.f, 0.f};
    #pragma unroll
    for (int k0 = 0; k0 < D; k0 += 4) {
        v2f a, b;
        a.x = Arow[k0];
        a.y = Arow[k0 + 1];
        b.x = Bcol[(size_t)k0 * D];
        b.y = Bcol[(size_t)(k0 + 1) * D];
        acc = __builtin_amdgcn_wmma_f32_16x16x4_f32(
            false, a, false, b, (short)0, acc, false, false);
    }

    // Epilogue: bias + eval-mode BN folded to scale/shift + ReLU (+ BN3 + ReLU).
    const int c = n0 + l15;
    const float bs = bias[c];
    float sc1 = 1.0f, sh1 = 0.0f, sc3 = 1.0f, sh3 = 0.0f;
    if (mode >= 1) {
        float g = bn[c], bb = bn[D + c], mm = bn[2 * D + c], vv = bn[3 * D + c];
        sc1 = g * rsqrtf(vv + BN_EPS);
        sh1 = bb - mm * sc1;
    }
    if (mode == 2) {
        float g = bn3[c], bb = bn3[D + c], mm = bn3[2 * D + c], vv = bn3[3 * D + c];
        sc3 = g * rsqrtf(vv + BN_EPS);
        sh3 = bb - mm * sc3;
    }

    float* out = Out + (size_t)m0 * D + c;
    #pragma unroll
    for (int r = 0; r < 8; ++r) {               // C layout: vgpr r -> row r + 8*half
        float v = acc[r] + bs;
        if (mode >= 1) v = fmaf(v, sc1, sh1);
        v = fmaxf(v, 0.0f);
        if (mode == 2) v = fmaxf(fmaf(v, sc3, sh3), 0.0f);
        out[(size_t)(r + 8 * half) * D] = v;
    }
}

// agg = (1 + eps[layer]) * h   (fuses the GIN self-term; scatter adds on top)
__global__ __launch_bounds__(256) void gin_init_agg(
    const float* __restrict__ h, float* __restrict__ agg,
    const float* __restrict__ eps, int layer)
{
    const float s = 1.0f + eps[layer];
    const int i = blockIdx.x * 256 + threadIdx.x;   // float4 index, exact grid
    const float4 v = ((const float4*)h)[i];
    ((float4*)agg)[i] = make_float4(s * v.x, s * v.y, s * v.z, s * v.w);
}

// One wave per edge: lane handles 4 of the 128 features (float4 load + 4 f32 atomics).
__global__ __launch_bounds__(256) void gin_scatter_add(
    const float* __restrict__ h, const long long* __restrict__ ei,
    float* __restrict__ agg)
{
    const int t    = blockIdx.x * 256 + threadIdx.x;
    const int e    = t >> 5;
    const int lane = t & 31;
    const long long s = ei[e];             // src row
    const long long d = ei[N_EDGES + e];   // dst row
    const float4 v = *(const float4*)(h + s * D + lane * 4);
    float* p = agg + d * D + lane * 4;
    atomicAdd(p + 0, v.x);
    atomicAdd(p + 1, v.y);
    atomicAdd(p + 2, v.z);
    atomicAdd(p + 3, v.w);
}

extern "C" void kernel_launch(void* const* d_in, const int* in_sizes, int n_in,
                              void* d_out, int out_size, void* d_ws, size_t ws_size,
                              hipStream_t stream)
{
    (void)in_sizes; (void)n_in; (void)out_size; (void)ws_size;

    const float*     x     = (const float*)d_in[0];
    const long long* ei    = (const long long*)d_in[1];   // int64 [2, N_EDGES]
    const float*     enc_W = (const float*)d_in[2];
    const float*     enc_b = (const float*)d_in[3];
    const float*     W1    = (const float*)d_in[4];
    const float*     b1    = (const float*)d_in[5];
    const float*     bn1   = (const float*)d_in[6];
    const float*     W2    = (const float*)d_in[7];
    const float*     b2    = (const float*)d_in[8];
    const float*     bn2   = (const float*)d_in[9];
    const float*     bn3   = (const float*)d_in[10];
    const float*     eps   = (const float*)d_in[11];

    float* h   = (float*)d_out;   // node features live in the output buffer
    float* agg = (float*)d_ws;    // 50000*128 floats = 25.6 MB scratch

    const int gemm_blocks = N_NODES / 16;              // 3125 (exact)
    const int init_blocks = (N_NODES * D / 4) / 256;   // 6250 (exact)
    const int scat_blocks = (N_EDGES * 32) / 256;      // 100000 (exact)

    // h = relu(x @ enc_W + enc_b)
    gemm_bn_relu<<<gemm_blocks, 256, 0, stream>>>(x, enc_W, enc_b, nullptr, nullptr, h, 0);

    for (int i = 0; i < 3; ++i) {
        gin_init_agg<<<init_blocks, 256, 0, stream>>>(h, agg, eps, i);
        gin_scatter_add<<<scat_blocks, 256, 0, stream>>>(h, ei, agg);
        // z = relu(bn1(agg @ W1 + b1))   -- in-place on agg (block reads its rows to LDS first)
        gemm_bn_relu<<<gemm_blocks, 256, 0, stream>>>(
            agg, W1 + (size_t)i * D * D, b1 + (size_t)i * D,
            bn1 + (size_t)i * 4 * D, nullptr, agg, 1);
        // h = relu(bn3(relu(bn2(z @ W2 + b2))))
        gemm_bn_relu<<<gemm_blocks, 256, 0, stream>>>(
            agg, W2 + (size_t)i * D * D, b2 + (size_t)i * D,
            bn2 + (size_t)i * 4 * D, bn3 + (size_t)i * 4 * D, h, 2);
    }
}